// QCausalConv1D_24524263260375
// MI455X (gfx1250) — compile-verified
//
#include <hip/hip_runtime.h>

// QCausalConv1D for MI455X (gfx1250):
//   y = silu(depthwise_causal_conv(x*is, w*ws) + b*bs); out = clip(rne(y/os), -128, 127)
// Memory-bound (~0.27 flops/byte): optimized for the CDNA5 async data path:
//   global_load_async_to_lds_b128  ->  LDS row tile  ->  ds_load_b128 windows
//   -> 4 FMA/output + v_exp_f32 SiLU -> global_store_b128.

static constexpr int QB = 2;
static constexpr int QD = 4096;
static constexpr int QL = 4096;
static constexpr int QW = 4;

static constexpr int THREADS    = 256;            // 8 waves (wave32)
static constexpr int PER_THREAD = QL / THREADS;   // 16 outputs / thread
static constexpr int HALO       = 4;              // 4-int pad in front (3 causal zeros + 1 slack)

// Low 32 bits of a generic pointer into LDS are the LDS byte offset
// (ISA 10.2, LDS aperture).
__device__ __forceinline__ unsigned lds_byte_addr(const void* p) {
    return (unsigned)(size_t)p;
}

__global__ __launch_bounds__(THREADS)
void qcausal_dwconv_silu_kernel(const int* __restrict__ x,      // [B, D, L] int8 values in int32
                                const int* __restrict__ wgt,    // [D, W]
                                const int* __restrict__ bias,   // [D]
                                const float* __restrict__ p_is,
                                const float* __restrict__ p_ws,
                                const float* __restrict__ p_os,
                                const float* __restrict__ p_bs,
                                float* __restrict__ out)        // [B, D, L] (int8 values as float)
{
    __shared__ __align__(16) int lds_x[QL + HALO];

    const int row = blockIdx.x;        // 0 .. B*D-1
    const int d   = row & (QD - 1);
    const int tid = threadIdx.x;

    // Zero the causal halo (lds_x[1..3] are read as x[-3..-1]; lds_x[0] is slack).
    if (tid < HALO) lds_x[tid] = 0;

    // ---- Async copy of this channel's row into LDS (ASYNCcnt path) ----
    // GVS addressing: mem = SGPR64 base + VGPR32 byte offset; VDST = LDS byte address.
    {
        const int* xrow = x + (size_t)row * QL;               // uniform -> SGPR pair
        const unsigned lds_base = lds_byte_addr(&lds_x[HALO]);
        #pragma unroll
        for (int p = 0; p < QL / (THREADS * 4); ++p) {        // 4 passes x 256 lanes x 16B = 16KB
            unsigned goff = (unsigned)((p * THREADS + tid) * 16);
            unsigned ldst = lds_base + goff;
            asm volatile("global_load_async_to_lds_b128 %0, %1, %2 offset:0"
                         :: "v"(ldst), "v"(goff), "s"(xrow)
                         : "memory");
        }
        asm volatile("s_wait_asynccnt 0" ::: "memory");       // my wave's copies landed in LDS
    }
    __syncthreads();                                          // all waves' copies visible

    // ---- Uniform per-channel constants ----
    const float is = p_is[0], ws = p_ws[0], os = p_os[0], bs = p_bs[0];
    const float sc     = is * ws;                             // fold both dequant scales
    const float inv_os = 1.0f / os;                           // one divide, hoisted out of the loop
    // Pre-scale taps by sc: inner loop becomes a pure 4-FMA chain.
    const float w0 = (float)wgt[d * QW + 0] * sc;
    const float w1 = (float)wgt[d * QW + 1] * sc;
    const float w2 = (float)wgt[d * QW + 2] * sc;
    const float w3 = (float)wgt[d * QW + 3] * sc;
    const float bf = (float)bias[d] * bs;

    // ---- Register-blocked compute: 16 outputs per thread ----
    const int l0 = tid * PER_THREAD;                          // multiple of 16 -> 64B LDS aligned

    // lds_x[l0 .. l0+19] covers x[l0-4 .. l0+15]; 5 aligned ds_load_b128.
    int v[PER_THREAD + HALO];
    #pragma unroll
    for (int j = 0; j < (PER_THREAD + HALO) / 4; ++j) {
        int4 t = ((const int4*)(&lds_x[l0]))[j];
        v[4 * j + 0] = t.x; v[4 * j + 1] = t.y;
        v[4 * j + 2] = t.z; v[4 * j + 3] = t.w;
    }

    // Convert window once (each value reused by up to 4 outputs).
    float vf[PER_THREAD + HALO];
    #pragma unroll
    for (int j = 0; j < PER_THREAD + HALO; ++j) vf[j] = (float)v[j];

    float4 res[PER_THREAD / 4];
    float* rv = (float*)res;
    #pragma unroll
    for (int i = 0; i < PER_THREAD; ++i) {
        // y[l0+i] = sum_k w[k]*sc * x[l0+i+k-3] + bf;  vf[j] = x[l0+j-4] -> index i+k+1
        float y = fmaf(w0, vf[i + 1], bf);
        y = fmaf(w1, vf[i + 2], y);
        y = fmaf(w2, vf[i + 3], y);
        y = fmaf(w3, vf[i + 4], y);
        float sg = 1.0f / (1.0f + __expf(-y));                // v_exp_f32 path
        y = y * sg;                                           // SiLU
        float q = rintf(y * inv_os);                          // round-half-even like jnp.round
        q = fminf(fmaxf(q, -128.0f), 127.0f);
        rv[i] = q;                                            // int8 value, stored as float
    }

    // ---- b128 stores ----
    float4* orow = (float4*)(out + (size_t)row * QL + l0);
    #pragma unroll
    for (int j = 0; j < PER_THREAD / 4; ++j) orow[j] = res[j];
}

extern "C" void kernel_launch(void* const* d_in, const int* in_sizes, int n_in,
                              void* d_out, int out_size, void* d_ws, size_t ws_size,
                              hipStream_t stream) {
    (void)in_sizes; (void)n_in; (void)d_ws; (void)ws_size; (void)out_size;
    const int*   x   = (const int*)d_in[0];
    const int*   w   = (const int*)d_in[1];
    const int*   b   = (const int*)d_in[2];
    const float* is  = (const float*)d_in[3];
    const float* ws2 = (const float*)d_in[4];
    const float* os  = (const float*)d_in[5];
    const float* bs  = (const float*)d_in[6];
    float* out = (float*)d_out;

    dim3 grid(QB * QD);   // one block per (batch, channel) row
    dim3 block(THREADS);
    qcausal_dwconv_silu_kernel<<<grid, block, 0, stream>>>(x, w, b, is, ws2, os, bs, out);
}